// PointNet_FP_Module_49572512530939
// MI455X (gfx1250) — compile-verified
//
#include <hip/hip_runtime.h>

typedef float v2f __attribute__((ext_vector_type(2)));
typedef float v8f __attribute__((ext_vector_type(8)));

#define B_SZ    4
#define N_PTS   8192
#define M_PTS   2048
#define J_TOT   32768          // B*N
#define C_IN0   384
#define C_OUT0  256
#define C_OUT1  128

// ---------------------------------------------------------------------------
// Kernel 1: brute-force 3-NN + distance/normal weights
// grid = B * (N/256) = 128 blocks, 256 threads
// ---------------------------------------------------------------------------
__global__ void knn_weights_kernel(const float* __restrict__ xyz1,
                                   const float* __restrict__ xyz2,
                                   const float* __restrict__ norm1,
                                   const float* __restrict__ norm2,
                                   int*   __restrict__ idx_out,
                                   float* __restrict__ w_out) {
    __shared__ float px[M_PTS];
    __shared__ float py[M_PTS];
    __shared__ float pz[M_PTS];

    const int b = blockIdx.x >> 5;                       // 32 blocks per batch
    const int n = ((blockIdx.x & 31) << 8) + threadIdx.x;

    const float* x2 = xyz2 + (size_t)b * 3 * M_PTS;
    for (int i = threadIdx.x; i < M_PTS; i += 256) {
        px[i] = x2[i];
        py[i] = x2[M_PTS + i];
        pz[i] = x2[2 * M_PTS + i];
    }
    __syncthreads();

    const float qx = xyz1[(size_t)b * 3 * N_PTS + n];
    const float qy = xyz1[(size_t)b * 3 * N_PTS + N_PTS + n];
    const float qz = xyz1[(size_t)b * 3 * N_PTS + 2 * N_PTS + n];

    float d0 = 1e30f, d1 = 1e30f, d2 = 1e30f;
    int   i0 = 0, i1 = 0, i2 = 0;
    for (int m = 0; m < M_PTS; ++m) {
        const float dx = qx - px[m];
        const float dy = qy - py[m];
        const float dz = qz - pz[m];
        const float d  = dx * dx + dy * dy + dz * dz;
        if (d < d2) {
            if (d < d1) {
                d2 = d1; i2 = i1;
                if (d < d0) { d1 = d0; i1 = i0; d0 = d; i0 = m; }
                else        { d1 = d;  i1 = m; }
            } else { d2 = d; i2 = m; }
        }
    }

    const float nqx = norm1[(size_t)b * 3 * N_PTS + n];
    const float nqy = norm1[(size_t)b * 3 * N_PTS + N_PTS + n];
    const float nqz = norm1[(size_t)b * 3 * N_PTS + 2 * N_PTS + n];
    const float* n2 = norm2 + (size_t)b * 3 * M_PTS;

    int   ii[3] = { i0, i1, i2 };
    float dd[3] = { d0, d1, d2 };
    float rw[3], rn[3];
    float sw = 0.0f, sn = 0.0f;
#pragma unroll
    for (int k = 0; k < 3; ++k) {
        const float dist = sqrtf(fmaxf(dd[k], 1e-20f));
        const float rd   = 1.0f / fmaxf(dist, 1e-10f);
        rw[k] = rd; sw += rd;
        const float ex = nqx - n2[ii[k]];
        const float ey = nqy - n2[M_PTS + ii[k]];
        const float ez = nqz - n2[2 * M_PTS + ii[k]];
        const float nd = sqrtf(ex * ex + ey * ey + ez * ez);
        const float rr = 1.0f / fmaxf(nd, 1e-10f);
        rn[k] = rr; sn += rr;
    }
    const int base = ((b * N_PTS) + n) * 3;
#pragma unroll
    for (int k = 0; k < 3; ++k) {
        w_out[base + k]   = (rw[k] / sw) * (rn[k] / sn);
        idx_out[base + k] = ii[k];
    }
}

// ---------------------------------------------------------------------------
// Kernel 2: three_interpolate + concat -> X (384 x 32768), row-major [c][j]
// grid = dim3(J/256, 384)
// ---------------------------------------------------------------------------
__global__ void interp_concat_kernel(const float* __restrict__ points1,
                                     const float* __restrict__ points2,
                                     const int*   __restrict__ idx,
                                     const float* __restrict__ w,
                                     float* __restrict__ X) {
    const int c = blockIdx.y;
    const int j = blockIdx.x * 256 + threadIdx.x;
    const int b = j >> 13;          // / 8192
    const int n = j & 8191;
    float v;
    if (c < C_OUT0) {
        const float* p2 = points2 + ((size_t)b * C_OUT0 + c) * M_PTS;
        const int base  = ((b * N_PTS) + n) * 3;
        v = w[base + 0] * p2[idx[base + 0]]
          + w[base + 1] * p2[idx[base + 1]]
          + w[base + 2] * p2[idx[base + 2]];
    } else {
        v = points1[((size_t)b * C_OUT1 + (c - C_OUT0)) * N_PTS + n];
    }
    X[(size_t)c * J_TOT + j] = v;
}

// ---------------------------------------------------------------------------
// Kernel 3/5: WMMA fp32 GEMM  Y[o][j] = sum_k W[o][k] * f(Xin[k][j]) + bias[o]
//   f = identity (scale==nullptr) or relu(scale[k]*x + shift[k]) (fused BN)
// Block: 256 threads (8 waves). Block tile: 64(O) x 128(J). Wave tile: 16x64.
// grid = dim3(J/128, Otot/64)
// Uses V_WMMA_F32_16X16X4_F32; layouts per CDNA5 ISA 7.12.2.
// ---------------------------------------------------------------------------
__global__ void gemm_wmma_kernel(const float* __restrict__ Wm,
                                 const float* __restrict__ Xin,
                                 const float* __restrict__ bias,
                                 const float* __restrict__ scale,   // nullable
                                 const float* __restrict__ shift,   // nullable
                                 float* __restrict__ Y,
                                 int Ktot) {
    // LDS: K-slab of X, stored transposed [j][k], stride 34 (conflict-free b64)
    __shared__ float xs[128][34];

    const int lane = threadIdx.x & 31;
    const int wave = threadIdx.x >> 5;
    const int wo   = wave & 3;        // 4 O sub-tiles of 16
    const int wj   = wave >> 2;       // 2 J sub-tiles of 64
    const int half = lane >> 4;       // 0: K=0,1 ; 1: K=2,3 (A); same for B
    const int lrow = lane & 15;

    const int j_base = blockIdx.x * 128;
    const int o_base = blockIdx.y * 64;
    const int o_row  = o_base + wo * 16 + lrow;     // A-fragment row

    v8f acc0 = {}, acc1 = {}, acc2 = {}, acc3 = {};

    for (int ks = 0; ks < Ktot; ks += 32) {
        // ---- stage 32 x 128 slab of Xin into LDS (transposed), fused BN/relu
        {
            const bool bn = (scale != nullptr);
#pragma unroll
            for (int i = 0; i < 16; ++i) {
                const int e  = threadIdx.x + i * 256;  // 0..4095
                const int kk = e >> 7;                 // /128
                const int jj = e & 127;
                const int kg = ks + kk;
                float v = Xin[(size_t)kg * J_TOT + j_base + jj];
                if (bn) v = fmaxf(0.0f, v * scale[kg] + shift[kg]);
                xs[jj][kk] = v;
            }
            // prefetch next slab (speculative, CDNA5 global_prefetch)
            if (ks + 32 < Ktot)
                __builtin_prefetch(&Xin[(size_t)(ks + 32) * J_TOT + j_base + threadIdx.x], 0, 1);
        }
        __syncthreads();

        // ---- 8 K-steps of 4
#pragma unroll
        for (int k4 = 0; k4 < 32; k4 += 4) {
            const int kg = ks + k4 + half * 2;
            const v2f a = *(const v2f*)(Wm + (size_t)o_row * Ktot + kg);

            const int jl0 = wj * 64 + lrow;
            const v2f bf0 = *(const v2f*)&xs[jl0 +  0][k4 + half * 2];
            const v2f bf1 = *(const v2f*)&xs[jl0 + 16][k4 + half * 2];
            const v2f bf2 = *(const v2f*)&xs[jl0 + 32][k4 + half * 2];
            const v2f bf3 = *(const v2f*)&xs[jl0 + 48][k4 + half * 2];

            acc0 = __builtin_amdgcn_wmma_f32_16x16x4_f32(false, a, false, bf0,
                                                         (short)0, acc0, false, false);
            acc1 = __builtin_amdgcn_wmma_f32_16x16x4_f32(false, a, false, bf1,
                                                         (short)0, acc1, false, false);
            acc2 = __builtin_amdgcn_wmma_f32_16x16x4_f32(false, a, false, bf2,
                                                         (short)0, acc2, false, false);
            acc3 = __builtin_amdgcn_wmma_f32_16x16x4_f32(false, a, false, bf3,
                                                         (short)0, acc3, false, false);
        }
        __syncthreads();
    }

    // ---- epilogue: D layout: VGPR v -> M = v + half*8, N = lrow
    v8f accs[4] = { acc0, acc1, acc2, acc3 };
#pragma unroll
    for (int nt = 0; nt < 4; ++nt) {
        const int j = j_base + wj * 64 + nt * 16 + lrow;
#pragma unroll
        for (int v = 0; v < 8; ++v) {
            const int o = o_base + wo * 16 + half * 8 + v;
            Y[(size_t)o * J_TOT + j] = accs[nt][v] + bias[o];
        }
    }
}

// ---------------------------------------------------------------------------
// Kernel 4a: per-channel sum / sumsq partial reduction
// grid = dim3(32, C); each block reduces 1024 j's of one channel
// ---------------------------------------------------------------------------
__global__ void bn_stats_kernel(const float* __restrict__ Y,
                                float* __restrict__ sums,
                                float* __restrict__ sumsq) {
    __shared__ float ls[256];
    __shared__ float lq[256];
    const int c  = blockIdx.y;
    const int j0 = blockIdx.x * 1024 + threadIdx.x;
    float s = 0.0f, q = 0.0f;
#pragma unroll
    for (int i = 0; i < 4; ++i) {
        const float v = Y[(size_t)c * J_TOT + j0 + i * 256];
        s += v; q += v * v;
    }
    ls[threadIdx.x] = s;
    lq[threadIdx.x] = q;
    __syncthreads();
    for (int off = 128; off > 0; off >>= 1) {
        if (threadIdx.x < off) {
            ls[threadIdx.x] += ls[threadIdx.x + off];
            lq[threadIdx.x] += lq[threadIdx.x + off];
        }
        __syncthreads();
    }
    if (threadIdx.x == 0) {
        atomicAdd(&sums[c],  ls[0]);
        atomicAdd(&sumsq[c], lq[0]);
    }
}

// ---------------------------------------------------------------------------
// Kernel 4b: fold BN stats into scale/shift
// ---------------------------------------------------------------------------
__global__ void bn_finalize_kernel(const float* __restrict__ sums,
                                   const float* __restrict__ sumsq,
                                   const float* __restrict__ g,
                                   const float* __restrict__ beta,
                                   float* __restrict__ scale,
                                   float* __restrict__ shift,
                                   int C) {
    const int c = blockIdx.x * blockDim.x + threadIdx.x;
    if (c < C) {
        const float inv_n = 1.0f / (float)J_TOT;
        const float mean  = sums[c] * inv_n;
        const float var   = sumsq[c] * inv_n - mean * mean;
        const float inv   = rsqrtf(var + 1e-5f);
        const float sc    = g[c] * inv;
        scale[c] = sc;
        shift[c] = beta[c] - mean * sc;
    }
}

// ---------------------------------------------------------------------------
// Kernel 6: final BN + relu, relayout [o][b*N+n] -> out (B, 128, N)
// grid = 4*128*8192/256 = 16384
// ---------------------------------------------------------------------------
__global__ void apply_out_kernel(const float* __restrict__ Y1,
                                 const float* __restrict__ scale,
                                 const float* __restrict__ shift,
                                 float* __restrict__ out) {
    const int e = blockIdx.x * 256 + threadIdx.x;
    const int b = e >> 20;            // / (128*8192)
    const int r = e & ((1 << 20) - 1);
    const int o = r >> 13;            // / 8192
    const int n = r & 8191;
    const float v = Y1[(size_t)o * J_TOT + b * N_PTS + n];
    out[e] = fmaxf(0.0f, scale[o] * v + shift[o]);
}

// ---------------------------------------------------------------------------
// Host launcher
// ---------------------------------------------------------------------------
extern "C" void kernel_launch(void* const* d_in, const int* in_sizes, int n_in,
                              void* d_out, int out_size, void* d_ws, size_t ws_size,
                              hipStream_t stream) {
    const float* xyz1    = (const float*)d_in[0];
    const float* xyz2    = (const float*)d_in[1];
    const float* norm1   = (const float*)d_in[2];
    const float* norm2   = (const float*)d_in[3];
    const float* points1 = (const float*)d_in[4];
    const float* points2 = (const float*)d_in[5];
    const float* W0      = (const float*)d_in[6];
    const float* b0      = (const float*)d_in[7];
    const float* g0      = (const float*)d_in[8];
    const float* be0     = (const float*)d_in[9];
    const float* W1      = (const float*)d_in[10];
    const float* b1      = (const float*)d_in[11];
    const float* g1      = (const float*)d_in[12];
    const float* be1     = (const float*)d_in[13];

    float* ws = (float*)d_ws;
    // workspace layout (floats)
    float* X    = ws;                                   // 384*32768 = 12,582,912
    float* Y1   = ws;                                   // aliases X (X dead after GEMM0)
    float* Y0   = ws + 12582912;                        // 256*32768 = 8,388,608
    float* wgt  = ws + 12582912 + 8388608;              // 294,912
    int*   idx  = (int*)(wgt + 294912);                 // 294,912
    float* st   = (float*)(idx + 294912);               // stats block
    float* sums0  = st;          float* sumsq0 = st + 256;
    float* scale0 = st + 512;    float* shift0 = st + 768;
    float* sums1  = st + 1024;   float* sumsq1 = st + 1152;
    float* scale1 = st + 1280;   float* shift1 = st + 1408;

    // zero the stats accumulators (graph-capture safe)
    hipMemsetAsync(st, 0, 1536 * sizeof(float), stream);

    // 1) 3-NN + weights
    knn_weights_kernel<<<dim3(B_SZ * (N_PTS / 256)), 256, 0, stream>>>(
        xyz1, xyz2, norm1, norm2, idx, wgt);

    // 2) interpolate + concat -> X
    interp_concat_kernel<<<dim3(J_TOT / 256, C_IN0), 256, 0, stream>>>(
        points1, points2, idx, wgt, X);

    // 3) GEMM0: Y0 = W0 * X + b0
    gemm_wmma_kernel<<<dim3(J_TOT / 128, C_OUT0 / 64), 256, 0, stream>>>(
        W0, X, b0, nullptr, nullptr, Y0, C_IN0);

    // 4) BN0 stats + finalize
    bn_stats_kernel<<<dim3(32, C_OUT0), 256, 0, stream>>>(Y0, sums0, sumsq0);
    bn_finalize_kernel<<<1, 256, 0, stream>>>(sums0, sumsq0, g0, be0,
                                              scale0, shift0, C_OUT0);

    // 5) GEMM1: Y1 = W1 * relu(bn0(Y0)) + b1   (BN+relu fused into staging)
    gemm_wmma_kernel<<<dim3(J_TOT / 128, C_OUT1 / 64), 256, 0, stream>>>(
        W1, Y0, b1, scale0, shift0, Y1, C_OUT0);

    // 6) BN1 stats + finalize
    bn_stats_kernel<<<dim3(32, C_OUT1), 256, 0, stream>>>(Y1, sums1, sumsq1);
    bn_finalize_kernel<<<1, 256, 0, stream>>>(sums1, sumsq1, g1, be1,
                                              scale1, shift1, C_OUT1);

    // 7) final BN + relu into d_out (B, 128, N)
    apply_out_kernel<<<dim3((B_SZ * C_OUT1 * N_PTS) / 256), 256, 0, stream>>>(
        Y1, scale1, shift1, (float*)d_out);
}